// TextEncoder_4303557231103
// MI455X (gfx1250) — compile-verified
//
#include <hip/hip_runtime.h>
#include <hip/hip_bf16.h>
#include <math.h>

// ---------------------------------------------------------------------------
// Mamba (2-layer) forward for MI455X / gfx1250, wave32.
// GEMMs: LDS-block-tiled V_WMMA_F32_16X16X4_F32 (fp32, matches fp32 ref),
// panels staged with GLOBAL_LOAD_ASYNC_TO_LDS (ASYNCcnt protocol).
// Selective scan parallelized over (batch, d_inner) channels.
// ---------------------------------------------------------------------------

#define N_LAYER 2
#define D_MODEL 256
#define D_INNER 512
#define D_STATE 16
#define DT_RANK 16
#define D_CONV  4
#define BATCH   4
#define SEQLEN  2048
#define MROWS   (BATCH * SEQLEN)   // 8192

// GEMM block tiling: 256 threads = 8 waves = 2(M) x 4(N) tiles of 16x16.
#define BM 32
#define BN 64
#define KT 16

typedef __attribute__((ext_vector_type(2))) float v2f;
typedef __attribute__((ext_vector_type(8))) float v8f;

#define AS1 __attribute__((address_space(1)))
#define AS3 __attribute__((address_space(3)))

__device__ __forceinline__ float silu_f(float x) {
    return x / (1.0f + expf(-x));
}

// --- gfx1250 async global->LDS copy (ASYNCcnt), with safe fallback ---------
__device__ __forceinline__ void async_copy_f32(const float* g, float* l) {
#if __has_builtin(__builtin_amdgcn_global_load_async_to_lds_b32)
    float* gnc = const_cast<float*>(g);
    __builtin_amdgcn_global_load_async_to_lds_b32(
        (AS1 int*)gnc, (AS3 int*)l, /*offset=*/0, /*cpol=*/0);
#else
    *l = *g;
#endif
}

__device__ __forceinline__ void async_wait_all() {
#if __has_builtin(__builtin_amdgcn_s_wait_asynccnt)
    __builtin_amdgcn_s_wait_asynccnt(0);
#elif __has_builtin(__builtin_amdgcn_global_load_async_to_lds_b32)
    asm volatile("s_wait_asynccnt 0x0" ::: "memory");
#endif
}

// ---------------------------------------------------------------------------
// Embedding gather: x[m, c] = emb[ids[m], c]
// ---------------------------------------------------------------------------
__global__ void embed_kernel(const int* __restrict__ ids,
                             const float* __restrict__ emb,
                             float* __restrict__ x, int total) {
    int i = blockIdx.x * blockDim.x + threadIdx.x;
    if (i >= total) return;
    int c = i & (D_MODEL - 1);
    int m = i >> 8;            // D_MODEL == 256
    x[i] = emb[(size_t)ids[m] * D_MODEL + c];
}

// ---------------------------------------------------------------------------
// LDS-tiled fp32 WMMA GEMM: C[M,N] = A[M,K] @ W[K,N]  (row-major, strides).
// Block computes BM x BN; K advances in KT chunks staged via async-to-LDS.
// Requires: M % BM == 0, K % KT == 0. N may be ragged (per-lane guards).
//
// WMMA fragment mapping (wave32, V_WMMA_F32_16X16X4_F32):
//   A (16x4): lanes 0-15: M=lane, VGPR0/1 = K+0/K+1; lanes 16-31: K+2/K+3.
//   B (4x16): lane&15 = N column, lane>>4 selects K pair (mirrored).
//   C/D     : VGPR r -> M = r + 8*(lane>>4), N = lane&15.
//
// EP==0: plain store.  EP==1: v = softplus(v + bias[n]) (dt_proj epilogue).
// ---------------------------------------------------------------------------
template <int EP>
__global__ void wmma_gemm_lds(const float* __restrict__ A, int lda,
                              const float* __restrict__ W, int ldw,
                              const float* __restrict__ bias,
                              float* __restrict__ C, int ldc,
                              int M, int N, int K) {
    __shared__ float sA[BM * KT];   // [m][k], row stride KT
    __shared__ float sB[KT * BN];   // [k][n], row stride BN

    const int tid  = threadIdx.x;
    const int lane = tid & 31;
    const int wave = tid >> 5;      // 0..7
    const int wm   = wave >> 2;     // 0..1  (M tile within block)
    const int wn   = wave & 3;      // 0..3  (N tile within block)

    const int tilesNb = (N + BN - 1) / BN;
    const int bm0 = (blockIdx.x / tilesNb) * BM;
    const int bn0 = (blockIdx.x % tilesNb) * BN;

    const int lo = lane & 15;
    const int hi = lane >> 4;
    const int kb = hi << 1;

    v8f acc = {};
    for (int k0 = 0; k0 < K; k0 += KT) {
        __syncthreads();   // previous chunk's compute done before restaging

        // stage A panel: BM x KT = 512 floats, 2 per thread (coalesced rows)
#pragma unroll
        for (int e = tid; e < BM * KT; e += 256) {
            int r = e >> 4;          // KT == 16
            int c = e & (KT - 1);
            async_copy_f32(A + (size_t)(bm0 + r) * lda + k0 + c, &sA[e]);
        }
        // stage B panel: KT x BN = 1024 floats, 4 per thread
#pragma unroll
        for (int e = tid; e < KT * BN; e += 256) {
            int r = e >> 6;          // BN == 64
            int c = e & (BN - 1);
            int n = bn0 + c;
            if (n < N) async_copy_f32(W + (size_t)(k0 + r) * ldw + n, &sB[e]);
            else       sB[e] = 0.0f;
        }
        async_wait_all();
        __syncthreads();

        const float* Arow = &sA[(wm * 16 + lo) * KT];
        const float* Bcol = &sB[wn * 16 + lo];
#pragma unroll
        for (int k = 0; k < KT; k += 4) {
            v2f a;
            a.x = Arow[k + kb];
            a.y = Arow[k + kb + 1];
            v2f b;
            b.x = Bcol[(k + kb) * BN];
            b.y = Bcol[(k + kb + 1) * BN];
            acc = __builtin_amdgcn_wmma_f32_16x16x4_f32(
                /*neg_a=*/false, a, /*neg_b=*/false, b,
                /*c_mod=*/(short)0, acc, /*reuse_a=*/false, /*reuse_b=*/false);
        }
    }

    // epilogue + store (per-lane N guard is after all WMMAs: EXEC safe)
    const int n = bn0 + wn * 16 + lo;
    if (n < N) {
        float bcol = 0.0f;
        if (EP == 1) bcol = bias[n];
#pragma unroll
        for (int r = 0; r < 8; ++r) {
            int m = bm0 + wm * 16 + r + (hi << 3);
            float v = acc[r];
            if (EP == 1) {
                v += bcol;
                v = (v > 20.0f) ? v : log1pf(expf(v));   // softplus
            }
            C[(size_t)m * ldc + n] = v;
        }
    }
}

// ---------------------------------------------------------------------------
// Depthwise causal conv (D_CONV=4, left pad 3) + bias + SiLU.
// Reads the u-half of xz [M, 2*D_INNER]; writes u_act [M, D_INNER].
// ---------------------------------------------------------------------------
__global__ void conv_silu_kernel(const float* __restrict__ xz,
                                 const float* __restrict__ wc,   // [D_INNER, D_CONV]
                                 const float* __restrict__ bc,   // [D_INNER]
                                 float* __restrict__ u_act, int total) {
    int i = blockIdx.x * blockDim.x + threadIdx.x;
    if (i >= total) return;
    int d = i & (D_INNER - 1);
    int m = i >> 9;                 // D_INNER == 512
    int l = m & (SEQLEN - 1);       // position within its batch row
    float s = bc[d];
#pragma unroll
    for (int k = 0; k < D_CONV; ++k) {
        int ls = l - (D_CONV - 1) + k;
        if (ls >= 0) {
            s += xz[(size_t)(m - (D_CONV - 1) + k) * (2 * D_INNER) + d] *
                 wc[d * D_CONV + k];
        }
    }
    u_act[i] = silu_f(s);
}

// ---------------------------------------------------------------------------
// Selective scan + D skip + SiLU(res) gating.
// grid = (D_INNER/256, BATCH), block = 256. One thread per (b, d) channel;
// h[16] state in VGPRs; B_t / C_t staged in LDS (32 floats) per timestep.
// ---------------------------------------------------------------------------
__global__ void scan_kernel(const float* __restrict__ u_act,   // [M, D_INNER]
                            const float* __restrict__ delta,   // [M, D_INNER]
                            const float* __restrict__ xdbl,    // [M, 48]
                            const float* __restrict__ xz,      // [M, 2*D_INNER]
                            const float* __restrict__ A_log,   // [D_INNER, D_STATE] (layer slice)
                            const float* __restrict__ Dp,      // [D_INNER]         (layer slice)
                            float* __restrict__ yg) {          // [M, D_INNER]
    __shared__ float sBC[2 * D_STATE];
    const int b = blockIdx.y;
    const int d = blockIdx.x * blockDim.x + threadIdx.x;

    float Areg[D_STATE];
    float h[D_STATE];
#pragma unroll
    for (int n = 0; n < D_STATE; ++n) {
        Areg[n] = -expf(A_log[(size_t)d * D_STATE + n]);
        h[n] = 0.0f;
    }
    const float Dval = Dp[d];

    for (int l = 0; l < SEQLEN; ++l) {
        const size_t row = (size_t)b * SEQLEN + l;
        if (threadIdx.x < 2 * D_STATE)
            sBC[threadIdx.x] = xdbl[row * (DT_RANK + 2 * D_STATE) + DT_RANK + threadIdx.x];
        __syncthreads();

        const float dt = delta[row * D_INNER + d];
        const float ut = u_act[row * D_INNER + d];
        const float du = dt * ut;
        float y = 0.0f;
#pragma unroll
        for (int n = 0; n < D_STATE; ++n) {
            h[n] = expf(dt * Areg[n]) * h[n] + du * sBC[n];
            y += h[n] * sBC[D_STATE + n];
        }
        y += ut * Dval;
        const float r = xz[row * (2 * D_INNER) + D_INNER + d];
        y *= silu_f(r);
        yg[row * D_INNER + d] = y;
        __syncthreads();   // protect sBC before next timestep's load
    }
}

// ---------------------------------------------------------------------------
// Host-side orchestration
// ---------------------------------------------------------------------------
static inline int gemm_blocks(int M, int N) {
    int tilesNb = (N + BN - 1) / BN;
    return (M / BM) * tilesNb;
}

extern "C" void kernel_launch(void* const* d_in, const int* in_sizes, int n_in,
                              void* d_out, int out_size, void* d_ws, size_t ws_size,
                              hipStream_t stream) {
    const int*   ids    = (const int*)  d_in[0];
    const float* emb    = (const float*)d_in[1];
    const float* w_in   = (const float*)d_in[2];   // [L, 256, 1024]
    const float* w_conv = (const float*)d_in[3];   // [L, 512, 1, 4]
    const float* b_conv = (const float*)d_in[4];   // [L, 512]
    const float* w_xp   = (const float*)d_in[5];   // [L, 512, 48]
    const float* w_dt   = (const float*)d_in[6];   // [L, 16, 512]
    const float* b_dt   = (const float*)d_in[7];   // [L, 512]
    const float* A_log  = (const float*)d_in[8];   // [L, 512, 16]
    const float* Dp     = (const float*)d_in[9];   // [L, 512]
    const float* w_out  = (const float*)d_in[10];  // [L, 512, 256]
    float* out = (float*)d_out;

    // workspace layout (fp32)
    float* ws    = (float*)d_ws;
    float* x     = ws;                                  // [8192, 256]
    float* xz    = x     + (size_t)MROWS * D_MODEL;     // [8192, 1024]
    float* u_act = xz    + (size_t)MROWS * 2 * D_INNER; // [8192, 512]
    float* xdbl  = u_act + (size_t)MROWS * D_INNER;     // [8192, 48]
    float* delta = xdbl  + (size_t)MROWS * (DT_RANK + 2 * D_STATE); // [8192, 512]
    float* yg    = delta + (size_t)MROWS * D_INNER;     // [8192, 512]

    // 1. embedding
    {
        int total = MROWS * D_MODEL;
        embed_kernel<<<(total + 255) / 256, 256, 0, stream>>>(ids, emb, x, total);
    }

    for (int l = 0; l < N_LAYER; ++l) {
        const float* Wi = w_in  + (size_t)l * D_MODEL * 2 * D_INNER;
        const float* Wc = w_conv+ (size_t)l * D_INNER * D_CONV;
        const float* Bc = b_conv+ (size_t)l * D_INNER;
        const float* Wx = w_xp  + (size_t)l * D_INNER * (DT_RANK + 2 * D_STATE);
        const float* Wd = w_dt  + (size_t)l * DT_RANK * D_INNER;
        const float* Bd = b_dt  + (size_t)l * D_INNER;
        const float* Al = A_log + (size_t)l * D_INNER * D_STATE;
        const float* Dl = Dp    + (size_t)l * D_INNER;
        const float* Wo = w_out + (size_t)l * D_INNER * D_MODEL;

        // 2. in_proj: xz = x @ Wi   (M=8192, K=256, N=1024)
        wmma_gemm_lds<0><<<gemm_blocks(MROWS, 2 * D_INNER), 256, 0, stream>>>(
            x, D_MODEL, Wi, 2 * D_INNER, nullptr,
            xz, 2 * D_INNER, MROWS, 2 * D_INNER, D_MODEL);

        // 3. depthwise causal conv + SiLU -> u_act
        {
            int total = MROWS * D_INNER;
            conv_silu_kernel<<<(total + 255) / 256, 256, 0, stream>>>(
                xz, Wc, Bc, u_act, total);
        }

        // 4. x_proj: xdbl = u_act @ Wx   (M=8192, K=512, N=48; ragged N)
        wmma_gemm_lds<0><<<gemm_blocks(MROWS, DT_RANK + 2 * D_STATE), 256, 0, stream>>>(
            u_act, D_INNER, Wx, DT_RANK + 2 * D_STATE, nullptr,
            xdbl, DT_RANK + 2 * D_STATE, MROWS, DT_RANK + 2 * D_STATE, D_INNER);

        // 5. dt_proj + bias + softplus: delta = softplus(xdbl[:, :16] @ Wd + Bd)
        //    (M=8192, K=16, N=512; A row stride is 48)
        wmma_gemm_lds<1><<<gemm_blocks(MROWS, D_INNER), 256, 0, stream>>>(
            xdbl, DT_RANK + 2 * D_STATE, Wd, D_INNER, Bd,
            delta, D_INNER, MROWS, D_INNER, DT_RANK);

        // 6. selective scan + D skip + SiLU(res) gating -> yg
        {
            dim3 grid(D_INNER / 256, BATCH);
            scan_kernel<<<grid, 256, 0, stream>>>(
                u_act, delta, xdbl, xz, Al, Dl, yg);
        }

        // 7. out_proj: next_x = yg @ Wo   (M=8192, K=512, N=256)
        {
            float* dst = (l == N_LAYER - 1) ? out : x;
            wmma_gemm_lds<0><<<gemm_blocks(MROWS, D_MODEL), 256, 0, stream>>>(
                yg, D_INNER, Wo, D_MODEL, nullptr,
                dst, D_MODEL, MROWS, D_MODEL, D_INNER);
        }
    }
}